// LowPrecAttention_23175643529618
// MI455X (gfx1250) — compile-verified
//
#include <hip/hip_runtime.h>
#include <math.h>

typedef __attribute__((ext_vector_type(2))) float v2f;
typedef __attribute__((ext_vector_type(8))) float v8f;

#define M_BITS 10
#define SHIFT  (23 - M_BITS)   // 13

// Round-to-nearest-even of the fp32 mantissa to M_BITS bits.
// Equivalent to the reference frexp/round/ldexp for finite values
// (carry out of the mantissa correctly bumps the exponent).
__device__ __forceinline__ float round_m10(float x) {
    unsigned u = __float_as_uint(x);
    u += ((u >> SHIFT) & 1u) + ((1u << (SHIFT - 1)) - 1u);
    u &= ~((1u << SHIFT) - 1u);
    return __uint_as_float(u);
}

__device__ __forceinline__ v8f round_v8(v8f c) {
#pragma unroll
    for (int i = 0; i < 8; ++i) c[i] = round_m10(c[i]);
    return c;
}

// fp32 WMMA: D = A(16x4) * B(4x16) + C(16x16), full fp32 accumulate.
__device__ __forceinline__ v8f wmma4(v2f a, v2f b, v8f c) {
    return __builtin_amdgcn_wmma_f32_16x16x4_f32(
        /*neg_a=*/false, a, /*neg_b=*/false, b,
        /*c_mod=*/(short)0, c, /*reuse_a=*/false, /*reuse_b=*/false);
}

// ---------------------------------------------------------------------------
// Blocked-accumulation GEMM:  out = round10_blocked(A @ W^T) ; then +bias,
// round again.  A:[M,K] row-major, W:[N,K] row-major (so B[k][n]=W[n][k]).
// Each wave owns a 16x64 strip: 1 A-fragment feeds 4 wmmas per K-step.
// mode 0: plain [M,N] store (output projection)
// mode 1: QKV scatter into q/k/v buffers laid out [B,H,T,D]
// ---------------------------------------------------------------------------
__global__ __launch_bounds__(128) void gemm_blk_kernel(
    const float* __restrict__ A, const float* __restrict__ W,
    const float* __restrict__ bias, float* __restrict__ outP,
    float* __restrict__ qb, float* __restrict__ kb, float* __restrict__ vb,
    int M, int N, int K, int mode) {
    const int lane = threadIdx.x & 31;
    const int wave = threadIdx.x >> 5;
    const int l16  = lane & 15;
    const int hi   = lane >> 4;          // 0: K even-pair/rows 0-7, 1: +2 / rows 8-15

    const int  ngrp  = N >> 6;           // groups of four 16-col tiles
    const long tile  = (long)blockIdx.x * 4 + wave;
    const long total = (long)(M >> 4) * ngrp;
    if (tile >= total) return;
    const int m0 = (int)(tile / ngrp) << 4;
    const int n0 = (int)(tile % ngrp) << 6;

    // A fragment (16x4 fp32): lane l16 holds row m0+l16, K pair (2*hi, 2*hi+1)
    const float* arow = A + (size_t)(m0 + l16) * K + 2 * hi;
    const float* w0   = W + (size_t)(n0 + l16) * K + 2 * hi;
    const float* w1   = w0 + (size_t)16 * K;
    const float* w2   = w0 + (size_t)32 * K;
    const float* w3   = w0 + (size_t)48 * K;

    v8f acc0 = {}, acc1 = {}, acc2 = {}, acc3 = {};
    for (int kblk = 0; kblk < K; kblk += 64) {
        if (kblk + 64 < K) {             // pull next K-block while computing
            __builtin_prefetch(arow + kblk + 64, 0, 1);
            __builtin_prefetch(w0 + kblk + 64, 0, 1);
        }
#pragma unroll
        for (int kk = 0; kk < 64; kk += 4) {
            const int kd = kblk + kk;
            v2f a = *(const v2f*)(arow + kd);
            acc0 = wmma4(a, *(const v2f*)(w0 + kd), acc0);
            acc1 = wmma4(a, *(const v2f*)(w1 + kd), acc1);
            acc2 = wmma4(a, *(const v2f*)(w2 + kd), acc2);
            acc3 = wmma4(a, *(const v2f*)(w3 + kd), acc3);
        }
        acc0 = round_v8(acc0);  acc1 = round_v8(acc1);
        acc2 = round_v8(acc2);  acc3 = round_v8(acc3);
    }

    v8f accs[4] = {acc0, acc1, acc2, acc3};
#pragma unroll
    for (int j = 0; j < 4; ++j) {
        const int   n  = n0 + j * 16 + l16;
        const float bv = bias[n];
        v8f c = accs[j];
#pragma unroll
        for (int r = 0; r < 8; ++r) {
            const int   m = m0 + r + 8 * hi;   // C/D layout: row r+8*hi, col l16
            const float v = round_m10(c[r] + bv);
            if (mode == 0) {
                outP[(size_t)m * N + n] = v;
            } else {
                const int which = n >> 11;          // 0:q 1:k 2:v
                const int cc = n & 2047;
                const int hh = cc >> 7, dd = cc & 127;
                const int bi = m >> 10, t = m & 1023;
                float* base = (which == 0) ? qb : ((which == 1) ? kb : vb);
                base[((size_t)(bi * 16 + hh) * 1024 + t) * 128 + dd] = v;
            }
        }
    }
}

// ---------------------------------------------------------------------------
// Attention: one wave per (b, h, 16-row tile).  Full 16x1024 masked score row
// lives in LDS (64 KB; CDNA5 WGP has 320 KB) so softmax is exact and the AV
// K-blocked rounding runs in the reference's fixed order.
// ---------------------------------------------------------------------------
__global__ __launch_bounds__(32) void attn_kernel(
    const float* __restrict__ qg, const float* __restrict__ kg,
    const float* __restrict__ vg, float* __restrict__ y) {
    __shared__ float satt[16 * 1024];        // 64 KB
    const int lane = threadIdx.x & 31;
    const int l16  = lane & 15;
    const int hi   = lane >> 4;

    const int rowTile = blockIdx.x & 63;
    const int bh      = blockIdx.x >> 6;     // b*16 + h
    const int b       = bh >> 4, h = bh & 15;
    const int m0      = rowTile * 16;

    const float* qh = qg + (size_t)bh * 1024 * 128;
    const float* kh = kg + (size_t)bh * 1024 * 128;
    const float* vh = vg + (size_t)bh * 1024 * 128;

    // Preload all q fragments (K = 0..127 -> 32 steps of 4) once per wave.
    v2f qa[32];
    {
        const float* qrow = qh + (size_t)(m0 + l16) * 128 + 2 * hi;
#pragma unroll
        for (int s = 0; s < 32; ++s) qa[s] = *(const v2f*)(qrow + 4 * s);
    }

    const float sqrtD = 11.313708498984761f;   // float(sqrt(128))

    // -------- scores = round_blk(q @ k^T), /sqrt(D), round, mask --------
    for (int nt = 0; nt < 64; ++nt) {
        const int n0 = nt * 16;
        const float* krow = kh + (size_t)(n0 + l16) * 128 + 2 * hi;
        v8f acc = {};
#pragma unroll
        for (int s = 0; s < 16; ++s)              // K-block 0..63
            acc = wmma4(qa[s], *(const v2f*)(krow + 4 * s), acc);
        acc = round_v8(acc);
#pragma unroll
        for (int s = 16; s < 32; ++s)             // K-block 64..127
            acc = wmma4(qa[s], *(const v2f*)(krow + 4 * s), acc);
        acc = round_v8(acc);

        const int col = n0 + l16;
#pragma unroll
        for (int r = 0; r < 8; ++r) {
            const int   rowg = m0 + r + 8 * hi;
            const float s    = round_m10(acc[r] / sqrtD);
            satt[(r + 8 * hi) * 1024 + col] =
                (col <= rowg) ? s : -__builtin_inff();
        }
    }

    // -------- exact softmax per row (single wave: LDS is in-order) --------
    for (int m = 0; m < 16; ++m) {
        float* row = satt + m * 1024;
        float mx = -__builtin_inff();
        for (int c = lane; c < 1024; c += 32) mx = fmaxf(mx, row[c]);
#pragma unroll
        for (int off = 16; off; off >>= 1) mx = fmaxf(mx, __shfl_xor(mx, off, 32));
        float sum = 0.0f;
        for (int c = lane; c < 1024; c += 32) {
            float e = expf(row[c] - mx);
            row[c] = e;
            sum += e;
        }
#pragma unroll
        for (int off = 16; off; off >>= 1) sum += __shfl_xor(sum, off, 32);
        const float inv = 1.0f / sum;
        for (int c = lane; c < 1024; c += 32) row[c] *= inv;
    }

    // -------- y = round_blk(att @ v), store transposed to [B,T,C] --------
    for (int dt = 0; dt < 8; ++dt) {
        const int d0 = dt * 16;
        v8f acc = {};
        for (int tb = 0; tb < 1024; tb += 64) {
#pragma unroll
            for (int kk = 0; kk < 64; kk += 4) {
                const int kd = tb + kk + 2 * hi;
                v2f a = *(const v2f*)(satt + l16 * 1024 + kd);   // ds_load_b64
                const float* vp = vh + (size_t)kd * 128 + d0 + l16;
                v2f bfr;
                bfr.x = vp[0];
                bfr.y = vp[128];
                acc = wmma4(a, bfr, acc);
            }
            acc = round_v8(acc);
        }
#pragma unroll
        for (int r = 0; r < 8; ++r) {
            const int t = m0 + r + 8 * hi;
            y[((size_t)(b * 1024 + t)) * 2048 + h * 128 + d0 + l16] = acc[r];
        }
    }
}

extern "C" void kernel_launch(void* const* d_in, const int* in_sizes, int n_in,
                              void* d_out, int out_size, void* d_ws, size_t ws_size,
                              hipStream_t stream) {
    const float* x      = (const float*)d_in[0];   // [4,1024,2048]
    const float* w_attn = (const float*)d_in[1];   // [6144,2048]
    const float* b_attn = (const float*)d_in[2];   // [6144]
    const float* w_proj = (const float*)d_in[3];   // [2048,2048]
    const float* b_proj = (const float*)d_in[4];   // [2048]
    float* out = (float*)d_out;                    // [4,1024,2048]

    const size_t Q = (size_t)4 * 16 * 1024 * 128;  // 8388608 floats per tensor
    float* q = (float*)d_ws;
    float* k = q + Q;
    float* v = k + Q;
    float* y = v + Q;

    // QKV projection: M=4096, N=6144, K=2048, scatter to [B,H,T,D]
    {
        const long waves  = (long)(4096 / 16) * (6144 / 64);   // 24576
        const int  blocks = (int)((waves + 3) / 4);
        gemm_blk_kernel<<<blocks, 128, 0, stream>>>(
            x, w_attn, b_attn, nullptr, q, k, v, 4096, 6144, 2048, 1);
    }

    // Attention: B*H*(T/16) = 4096 single-wave workgroups, 64 KB LDS each
    attn_kernel<<<4096, 32, 0, stream>>>(q, k, v, y);

    // Output projection: M=4096, N=2048, K=2048, plain store
    {
        const long waves  = (long)(4096 / 16) * (2048 / 64);   // 8192
        const int  blocks = (int)((waves + 3) / 4);
        gemm_blk_kernel<<<blocks, 128, 0, stream>>>(
            y, w_proj, b_proj, out, nullptr, nullptr, nullptr, 4096, 2048, 2048, 0);
    }
}